// SOM_4922032521526
// MI455X (gfx1250) — compile-verified
//
#include <hip/hip_runtime.h>
#include <math.h>

typedef float v2f __attribute__((ext_vector_type(2)));
typedef float v8f __attribute__((ext_vector_type(8)));

#define SOM_EPS   1e-6f
#define DIM       512
#define KCOLS     16384
#define BROWS     4096
#define BM        32          // rows of x per block
#define KT        128         // columns per block iteration (8 waves * 16)
#define XSTRIDE   516         // padded LDS row stride (floats): 4-float pad -> conflict-free

// ---------------------------------------------------------------------------
// Precompute per-column weight statistics: w2[k] = sum_d w[d][k]^2, sw[k] = sum_d w[d][k]
// ---------------------------------------------------------------------------
__global__ __launch_bounds__(256) void som_wstats_kernel(const float* __restrict__ w,
                                                         float* __restrict__ w2,
                                                         float* __restrict__ sw) {
    const int k = blockIdx.x * blockDim.x + threadIdx.x;
    if (k >= KCOLS) return;
    float s2 = 0.0f, s1 = 0.0f;
#pragma unroll 8
    for (int d = 0; d < DIM; ++d) {
        const float v = w[d * KCOLS + k];   // coalesced across threads
        s2 += v * v;
        s1 += v;
    }
    w2[k] = s2;
    sw[k] = s1;
}

// ---------------------------------------------------------------------------
// Fused SOM kernel: WMMA GEMM (x @ weight) + distance expansion + argmin +
// BMU location gather + mean-min-distance loss.
// Block: 256 threads = 8 wave32.  Grid: BROWS/BM = 128 blocks.
// ---------------------------------------------------------------------------
__global__ __launch_bounds__(256) void som_main_kernel(const float* __restrict__ x,
                                                       const float* __restrict__ w,
                                                       const float* __restrict__ loc,
                                                       const float* __restrict__ w2,
                                                       const float* __restrict__ sw,
                                                       float* __restrict__ out) {
    __shared__ float xs[BM * XSTRIDE];          // 32 x 512 x-tile, padded
    __shared__ float x2s[BM], sxs[BM];
    __shared__ float part2[BM][8], part1[BM][8];
    __shared__ float redv[8][BM];
    __shared__ unsigned redi[8][BM];

    const int tid  = threadIdx.x;
    const int wave = tid >> 5;
    const int lane = tid & 31;
    const int half = lane >> 4;     // 0: lanes 0-15, 1: lanes 16-31
    const int n    = lane & 15;     // column-in-tile for B/C, row-in-tile for A
    const int b0   = blockIdx.x * BM;

    // ---- stage the 32x512 x tile into LDS (float4 loads) ----
    for (int idx = tid; idx < BM * (DIM / 4); idx += 256) {
        const int row = idx >> 7;               // DIM/4 == 128
        const int dq  = (idx & 127) << 2;
        const float4 v = *(const float4*)(x + (size_t)(b0 + row) * DIM + dq);
        float* p = &xs[row * XSTRIDE + dq];
        p[0] = v.x; p[1] = v.y; p[2] = v.z; p[3] = v.w;
    }
    __syncthreads();

    // ---- per-row stats x2[b], sx[b] from LDS (8 partial chunks per row) ----
    {
        const int row = tid >> 3, ch = tid & 7;
        float s2 = 0.0f, s1 = 0.0f;
        const float* p = &xs[row * XSTRIDE + ch * 64];
#pragma unroll 8
        for (int i = 0; i < 64; ++i) { const float v = p[i]; s2 += v * v; s1 += v; }
        part2[row][ch] = s2;
        part1[row][ch] = s1;
    }
    __syncthreads();
    if (tid < BM) {
        float s2 = 0.0f, s1 = 0.0f;
#pragma unroll
        for (int ch = 0; ch < 8; ++ch) { s2 += part2[tid][ch]; s1 += part1[tid][ch]; }
        x2s[tid] = s2;
        sxs[tid] = s1;
    }
    __syncthreads();

    // lane-local running (min d2, argmin) for the 16 row-slots this lane owns
    float    best[2][8];
    unsigned bidx[2][8];
#pragma unroll
    for (int t = 0; t < 2; ++t)
#pragma unroll
        for (int r = 0; r < 8; ++r) { best[t][r] = 3.4e38f; bidx[t][r] = 0u; }

    const int koff = half * 2;      // A/B fragment K offset per ISA layout
    const float deps2 = (float)DIM * SOM_EPS * SOM_EPS;

    const float* paA = &xs[n * XSTRIDE];                 // A tile0: rows 0..15
    const float* paB = &xs[(16 + n) * XSTRIDE];          // A tile1: rows 16..31

    for (int kt = 0; kt < KCOLS; kt += KT) {
        const int col = kt + wave * 16 + n;
        const float w2c = w2[col];
        const float swc = sw[col];
        // prefetch next tile's weight column start into caches (global_prefetch_b8)
        if (kt + KT < KCOLS) __builtin_prefetch(w + (col + KT), 0, 1);

        v8f c0 = {};
        v8f c1 = {};
#pragma unroll 4
        for (int d0 = 0; d0 < DIM; d0 += 4) {
            const int dA = d0 + koff;
            v2f a0, a1, b;
            a0.x = paA[dA];     a0.y = paA[dA + 1];      // ds_load_b64
            a1.x = paB[dA];     a1.y = paB[dA + 1];
            b.x  = w[(size_t)dA * KCOLS + col];          // B 4x16, row-major weight
            b.y  = w[(size_t)(dA + 1) * KCOLS + col];
            c0 = __builtin_amdgcn_wmma_f32_16x16x4_f32(false, a0, false, b,
                                                       (short)0, c0, false, false);
            c1 = __builtin_amdgcn_wmma_f32_16x16x4_f32(false, a1, false, b,
                                                       (short)0, c1, false, false);
        }

        // d2 = x2 - 2*cross + w2 + 2eps*(sx - sw) + D*eps^2; track running min
        const float lin = w2c - 2.0f * SOM_EPS * swc + deps2;
#pragma unroll
        for (int r = 0; r < 8; ++r) {
            const int m0 = r + half * 8;                 // C layout: M = r + 8*half
            const float d2a = x2s[m0] + 2.0f * SOM_EPS * sxs[m0] - 2.0f * c0[r] + lin;
            if (d2a < best[0][r]) { best[0][r] = d2a; bidx[0][r] = (unsigned)col; }
            const int m1 = m0 + 16;
            const float d2b = x2s[m1] + 2.0f * SOM_EPS * sxs[m1] - 2.0f * c1[r] + lin;
            if (d2b < best[1][r]) { best[1][r] = d2b; bidx[1][r] = (unsigned)col; }
        }
    }

    // ---- cross-lane butterfly reduction within each 16-lane half ----
#pragma unroll
    for (int t = 0; t < 2; ++t)
#pragma unroll
        for (int r = 0; r < 8; ++r) {
            float v = best[t][r];
            unsigned i = bidx[t][r];
#pragma unroll
            for (int m = 8; m >= 1; m >>= 1) {
                const float    ov = __shfl_xor(v, m, 32);
                const unsigned oi = __shfl_xor(i, m, 32);
                if (ov < v || (ov == v && oi < i)) { v = ov; i = oi; }
            }
            best[t][r] = v;
            bidx[t][r] = i;
        }

    if (n == 0) {
#pragma unroll
        for (int t = 0; t < 2; ++t)
#pragma unroll
            for (int r = 0; r < 8; ++r) {
                const int row = t * 16 + half * 8 + r;
                redv[wave][row] = best[t][r];
                redi[wave][row] = bidx[t][r];
            }
    }
    __syncthreads();

    // ---- final reduction across the 8 waves + output ----
    if (tid < BM) {
        float v = redv[0][tid];
        unsigned i = redi[0][tid];
#pragma unroll
        for (int wv = 1; wv < 8; ++wv) {
            const float    ov = redv[wv][tid];
            const unsigned oi = redi[wv][tid];
            if (ov < v || (ov == v && oi < i)) { v = ov; i = oi; }
        }
        const int brow = b0 + tid;
        const float dist = sqrtf(fmaxf(v, 0.0f));
        out[2 * brow]     = loc[2 * i];         // BMU grid location (x)
        out[2 * brow + 1] = loc[2 * i + 1];     // BMU grid location (y)
        atomicAdd(out + 2 * BROWS, dist * (1.0f / (float)BROWS));
    }
}

// ---------------------------------------------------------------------------
extern "C" void kernel_launch(void* const* d_in, const int* in_sizes, int n_in,
                              void* d_out, int out_size, void* d_ws, size_t ws_size,
                              hipStream_t stream) {
    const float* x   = (const float*)d_in[0];   // (4096, 512)
    const float* w   = (const float*)d_in[1];   // (512, 16384)
    const float* loc = (const float*)d_in[2];   // (16384, 2)
    float* out = (float*)d_out;                 // 8192 location floats + 1 loss

    float* w2 = (float*)d_ws;                   // (16384,)
    float* sw = w2 + KCOLS;                     // (16384,)

    // zero the loss accumulator (graph-capturable)
    hipMemsetAsync(out + 2 * BROWS, 0, sizeof(float), stream);

    som_wstats_kernel<<<KCOLS / 256, 256, 0, stream>>>(w, w2, sw);

    som_main_kernel<<<BROWS / BM, 256, 0, stream>>>(x, w, loc, w2, sw, out);
}